// GRC_68599217652439
// MI455X (gfx1250) — compile-verified
//
#include <hip/hip_runtime.h>
#include <hip/hip_bf16.h>
#include <cstddef>
#include <cstdint>

// Problem constants (from reference)
#define NN   256   // state dim
#define MC   64    // control dim
#define PP   128   // output dim
#define HH   64    // history (window = HH+1 = 65)
#define TT   2048  // steps

typedef float v2f __attribute__((ext_vector_type(2)));
typedef float v8f __attribute__((ext_vector_type(8)));

// ---------------------------------------------------------------------------
// Async global->LDS 16B copy via CDNA5 GLOBAL_LOAD_ASYNC_TO_LDS_B128.
// The clang builtin wants CUDA-LangAS pointers that HIP can't spell, so we use
// inline asm (bridge-doc recommended path). VDST = per-lane LDS byte offset
// (= low 32 bits of generic shared pointer, ISA flat->LDS aperture rule),
// VADDR = 64-bit global address. Tracked by ASYNCcnt; joined with
// s_wait_asynccnt 0 before the tile barrier. Set USE_ASYNC_ASM 0 to fall back
// to plain b128 load + ds_store.
// ---------------------------------------------------------------------------
#define USE_ASYNC_ASM 1

__device__ __forceinline__ void copy16(const float* __restrict__ g, float* l) {
#if USE_ASYNC_ASM
  asm volatile("global_load_async_to_lds_b128 %0, %1, off"
               :: "v"((unsigned)(uintptr_t)l),
                  "v"((unsigned long long)(uintptr_t)g)
               : "memory");
#else
  *(float4*)l = *(const float4*)g;
#endif
}

__device__ __forceinline__ void async_join() {
#if USE_ASYNC_ASM
  asm volatile("s_wait_asynccnt 0x0" ::: "memory");
#endif
}

// ---------------------------------------------------------------------------
// f32 WMMA GEMM:  D[M x N] = A[M x K] * B[K x N] (+ Cin), all row-major.
// Block = 128 threads (4 waves). Block tile: 16(M) x 64(N). K-tiles of 32
// staged in LDS (A tile padded to stride 36 -> conflict-free fragment reads).
// Each wave owns one 16x16 tile, 8x V_WMMA_F32_16X16X4_F32 per K-tile.
// Grids are launched exact (no partial tiles) so no guards around barriers.
// ---------------------------------------------------------------------------
#define KT 32
#define ASTRIDE 36   // 36 floats = 144B: float4-aligned rows, banks 4L mod 64 distinct

__global__ void __launch_bounds__(128)
wmma_gemm_f32(const float* __restrict__ A, const float* __restrict__ Bm,
              const float* __restrict__ Cin, float* __restrict__ D,
              int M, int N, int K) {
  __shared__ float As[16 * ASTRIDE];
  __shared__ float Bs[KT * 64];
  const int tid  = threadIdx.x;
  const int lane = tid & 31;
  const int wave = tid >> 5;
  const int half = lane >> 4;
  const int l16  = lane & 15;
  const int n0b  = blockIdx.x * 64;
  const int m0   = blockIdx.y * 16;
  const int nw   = wave * 16;

  v8f acc = {0.f,0.f,0.f,0.f,0.f,0.f,0.f,0.f};
  if (Cin) {
#pragma unroll
    for (int r = 0; r < 8; ++r)
      acc[r] = Cin[(size_t)(m0 + r + 8 * half) * N + n0b + nw + l16];
  }

  for (int kt = 0; kt < K; kt += KT) {
    // A tile: 16 x 32 = 128 float4, one per thread
    {
      const int m = tid >> 3, q = tid & 7;
      copy16(&A[(size_t)(m0 + m) * K + kt + q * 4], &As[m * ASTRIDE + q * 4]);
    }
    // B tile: 32 x 64 = 512 float4, four per thread
#pragma unroll
    for (int j = 0; j < 4; ++j) {
      const int idx = tid + 128 * j;
      const int kk = idx >> 4, q = idx & 15;
      copy16(&Bm[(size_t)(kt + kk) * N + n0b + q * 4], &Bs[kk * 64 + q * 4]);
    }
    async_join();
    __syncthreads();
#pragma unroll
    for (int k0 = 0; k0 < KT; k0 += 4) {
      const int ka = k0 + 2 * half;
      v2f a, b;
      a[0] = As[l16 * ASTRIDE + ka];
      a[1] = As[l16 * ASTRIDE + ka + 1];
      b[0] = Bs[ka * 64 + nw + l16];
      b[1] = Bs[(ka + 1) * 64 + nw + l16];
      acc = __builtin_amdgcn_wmma_f32_16x16x4_f32(false, a, false, b,
                                                  (short)0, acc, false, false);
    }
    __syncthreads();
  }
#pragma unroll
  for (int r = 0; r < 8; ++r)
    D[(size_t)(m0 + r + 8 * half) * N + n0b + nw + l16] = acc[r];
}

// ---------------------------------------------------------------------------
// u_pert[t][m] = sum_{i=0}^{64} sum_p Mw[m][p][i] * ynat[t-i][p]   (ynat<0 = 0)
// Windowed GEMM, K = 65*128. grid = (1, TT/16), block = 128 (4 n-tiles of 16).
// ---------------------------------------------------------------------------
__global__ void __launch_bounds__(128)
upert_wmma(const float* __restrict__ Ynat, const float* __restrict__ Mw,
           float* __restrict__ Upert) {
  const int lane = threadIdx.x & 31;
  const int wave = threadIdx.x >> 5;
  const int half = lane >> 4;
  const int l16  = lane & 15;
  const int n0 = wave * 16;
  const int t0 = blockIdx.y * 16;

  v8f acc = {0.f,0.f,0.f,0.f,0.f,0.f,0.f,0.f};
  const int trow = t0 + l16;
  const size_t mwrow = (size_t)(n0 + l16) * (PP * (HH + 1));

  for (int i = 0; i <= HH; ++i) {
    const int src = trow - i;
    const float* yrow = (src >= 0) ? (Ynat + (size_t)src * PP) : nullptr;
    for (int pk = 0; pk < PP; pk += 4) {
      const int ka = pk + 2 * half;
      v2f a, b;
      a[0] = yrow ? yrow[ka]     : 0.0f;
      a[1] = yrow ? yrow[ka + 1] : 0.0f;
      b[0] = Mw[mwrow + (size_t)ka * (HH + 1) + i];
      b[1] = Mw[mwrow + (size_t)(ka + 1) * (HH + 1) + i];
      acc = __builtin_amdgcn_wmma_f32_16x16x4_f32(false, a, false, b,
                                                  (short)0, acc, false, false);
    }
  }
#pragma unroll
  for (int r = 0; r < 8; ++r)
    Upert[(size_t)(t0 + r + 8 * half) * MC + n0 + l16] = acc[r];
}

// ---------------------------------------------------------------------------
// Sequential scan x_t = M x_{t-1} + d_t, x_0 = 0. The 256x256 matrix (row-major,
// UNtransposed) is loaded ONCE into LDS (padded stride 260 -> aligned b128 row
// reads, 2-way bank conflict max), so the 2048-step loop never touches global
// for the matrix. d_t = Dmat[t,:] or w[t] broadcast. 1024 thr, 4-way split-K.
// Dynamic LDS: (256*260 + 256 + 1024) floats = 271360 B < 320 KB WGP LDS.
// ---------------------------------------------------------------------------
#define MSTRIDE 260
#define LINSCAN_LDS_BYTES ((NN * MSTRIDE + NN + 1024) * 4)

__global__ void __launch_bounds__(1024)
linscan_kernel(const float* __restrict__ Min, const float* __restrict__ Dmat,
               const float* __restrict__ w, float* __restrict__ Xout) {
  extern __shared__ float lds[];
  float* Ms   = lds;                 // NN * MSTRIDE
  float* xs   = Ms + NN * MSTRIDE;   // NN
  float* part = xs + NN;             // 1024
  const int tid = threadIdx.x;

  // one-time matrix stage: 256 rows x 64 float4
  for (int i = tid; i < NN * 64; i += 1024) {
    const int rr = i >> 6, q = i & 63;
    copy16(&Min[(size_t)rr * NN + q * 4], &Ms[rr * MSTRIDE + q * 4]);
  }
  async_join();

  const int r = tid & (NN - 1);
  const int s = tid >> 8;            // 0..3 k-segments of 64
  if (s == 0) { xs[r] = 0.0f; Xout[r] = 0.0f; }
  __syncthreads();

  const float* mrow = Ms + (size_t)r * MSTRIDE + s * 64;
  const float* xseg = xs + s * 64;
  for (int t = 1; t < TT; ++t) {
    float acc = 0.0f;
#pragma unroll
    for (int k = 0; k < 64; k += 4) {
      const float4 mv = *(const float4*)(mrow + k);
      acc += mv.x * xseg[k] + mv.y * xseg[k + 1] +
             mv.z * xseg[k + 2] + mv.w * xseg[k + 3];
    }
    part[tid] = acc;
    __syncthreads();
    if (s == 0) {
      float v = part[r] + part[r + 256] + part[r + 512] + part[r + 768];
      v += Dmat ? Dmat[(size_t)t * NN + r] : w[t];
      xs[r] = v;
      Xout[(size_t)t * NN + r] = v;
    }
    __syncthreads();
  }
}

// ---------------------------------------------------------------------------
// Small helpers
// ---------------------------------------------------------------------------
__global__ void transpose_scale(const float* __restrict__ in, float* __restrict__ out,
                                int rows, int cols, float scale) {
  int idx = blockIdx.x * blockDim.x + threadIdx.x;
  if (idx < rows * cols) {
    int rr = idx / cols, cc = idx % cols;
    out[(size_t)cc * rows + rr] = scale * in[idx];
  }
}

// Acl = A - Bmk (elementwise)
__global__ void sub_kernel(const float* __restrict__ a, const float* __restrict__ b,
                           float* __restrict__ out, int n) {
  int idx = blockIdx.x * blockDim.x + threadIdx.x;
  if (idx < n) out[idx] = a[idx] - b[idx];
}

// Vmat[t] = Upert[t] + bias ; Vprev[t] = Vmat[t-1] (Vprev[0] = 0)
__global__ void vmat_kernel(const float* __restrict__ Upert, const float* __restrict__ bias,
                            float* __restrict__ Vmat, float* __restrict__ Vprev) {
  int idx = blockIdx.x * blockDim.x + threadIdx.x;
  if (idx < TT * MC) {
    int t = idx / MC, m = idx % MC;
    Vmat[idx] = Upert[idx] + bias[m];
    Vprev[idx] = (t == 0) ? 0.0f : (Upert[idx - MC] + bias[m]);
  }
}

// Dmat[t,:] += w[t] for t>=1 ; Dmat[0,:] = 0 (x_0 = 0, w_eff(0)=0)
__global__ void addw_kernel(float* __restrict__ Dmat, const float* __restrict__ w) {
  int idx = blockIdx.x * blockDim.x + threadIdx.x;
  if (idx < TT * NN) {
    int t = idx >> 8;
    if (t >= 1) Dmat[idx] += w[t];
    else        Dmat[idx] = 0.0f;
  }
}

// cost_t = <Y_t, Z1_t> + <U_t, Z2_t>
__global__ void cost_kernel(const float* __restrict__ Y, const float* __restrict__ Z1,
                            const float* __restrict__ U, const float* __restrict__ Z2,
                            float* __restrict__ out) {
  int t = blockIdx.x * blockDim.x + threadIdx.x;
  if (t < TT) {
    float c = 0.0f;
    const float* y  = Y  + (size_t)t * PP;
    const float* z1 = Z1 + (size_t)t * PP;
#pragma unroll 4
    for (int p = 0; p < PP; ++p) c += y[p] * z1[p];
    const float* u  = U  + (size_t)t * MC;
    const float* z2 = Z2 + (size_t)t * MC;
#pragma unroll 4
    for (int m = 0; m < MC; ++m) c += u[m] * z2[m];
    out[t] = c;
  }
}

// ---------------------------------------------------------------------------
extern "C" void kernel_launch(void* const* d_in, const int* in_sizes, int n_in,
                              void* d_out, int out_size, void* d_ws, size_t ws_size,
                              hipStream_t stream) {
  const float* A    = (const float*)d_in[0];   // 256x256
  const float* B    = (const float*)d_in[1];   // 256x64
  const float* C    = (const float*)d_in[2];   // 128x256
  const float* Q    = (const float*)d_in[3];   // 128x128
  const float* R    = (const float*)d_in[4];   // 64x64
  const float* K    = (const float*)d_in[5];   // 64x128
  const float* Mw   = (const float*)d_in[6];   // 64x128x65
  const float* bias = (const float*)d_in[7];   // 64
  const float* w    = (const float*)d_in[8];   // 2048
  float* out = (float*)d_out;                  // 2048

  // allow 271KB dynamic LDS for the scan kernel (host-side attr, capture-safe)
  (void)hipFuncSetAttribute((const void*)linscan_kernel,
                            hipFuncAttributeMaxDynamicSharedMemorySize,
                            (int)LINSCAN_LDS_BYTES);

  // Workspace layout (floats)
  float* ws = (float*)d_ws;
  size_t off = 0;
  float* CT    = ws + off; off += NN * PP;           // C^T   (256x128)
  float* BT    = ws + off; off += MC * NN;           // B^T   (64x256)
  float* KTn   = ws + off; off += PP * MC;           // -K^T  (128x64)
  float* KC    = ws + off; off += MC * NN;           // K@C   (64x256)
  float* Bmk   = ws + off; off += NN * NN;           // B@(K@C)
  float* Acl   = ws + off; off += NN * NN;           // A - BKC
  float* Xw    = ws + off; off += (size_t)TT * NN;   // disturbance-only states
  float* Ynat  = ws + off; off += (size_t)TT * PP;   // y_nat_t = C x^w_t
  float* Upert = ws + off; off += (size_t)TT * MC;
  float* Vmat  = ws + off; off += (size_t)TT * MC;   // v_t = u_pert_t + bias
  float* Vprev = ws + off; off += (size_t)TT * MC;   // v_{t-1}
  float* Dmat  = ws + off; off += (size_t)TT * NN;   // B v_{t-1} + w_t
  float* X     = ws + off; off += (size_t)TT * NN;   // closed-loop states
  float* Ymat  = ws + off; off += (size_t)TT * PP;   // y_t
  float* Umat  = ws + off; off += (size_t)TT * MC;   // u_t
  float* Z1    = ws + off; off += (size_t)TT * PP;   // Y@Q
  float* Z2    = ws + off; off += (size_t)TT * MC;   // U@R
  (void)ws_size; (void)in_sizes; (void)n_in; (void)out_size;

  // --- Phase 1: transposes + closed-loop matrix Acl = A - B K C ---
  transpose_scale<<<dim3((PP*NN+255)/256), 256, 0, stream>>>(C, CT, PP, NN, 1.0f);
  transpose_scale<<<dim3((NN*MC+255)/256), 256, 0, stream>>>(B, BT, NN, MC, 1.0f);
  transpose_scale<<<dim3((MC*PP+255)/256), 256, 0, stream>>>(K, KTn, MC, PP, -1.0f);
  wmma_gemm_f32<<<dim3(NN/64, MC/16), 128, 0, stream>>>(K, C, nullptr, KC, MC, NN, PP);
  wmma_gemm_f32<<<dim3(NN/64, NN/16), 128, 0, stream>>>(B, KC, nullptr, Bmk, NN, NN, MC);
  sub_kernel<<<dim3((NN*NN+255)/256), 256, 0, stream>>>(A, Bmk, Acl, NN*NN);

  // --- Phase 2: disturbance scan x^w_t = A x^w_{t-1} + w_t ; Ynat = Xw @ C^T ---
  linscan_kernel<<<1, 1024, LINSCAN_LDS_BYTES, stream>>>(A, nullptr, w, Xw);
  wmma_gemm_f32<<<dim3(PP/64, TT/16), 128, 0, stream>>>(Xw, CT, nullptr, Ynat, TT, PP, NN);

  // --- Phase 3: u_pert, v_t, drive D_t = B v_{t-1} + w_t ---
  upert_wmma<<<dim3(1, TT/16), 128, 0, stream>>>(Ynat, Mw, Upert);
  vmat_kernel<<<dim3((TT*MC+255)/256), 256, 0, stream>>>(Upert, bias, Vmat, Vprev);
  wmma_gemm_f32<<<dim3(NN/64, TT/16), 128, 0, stream>>>(Vprev, BT, nullptr, Dmat, TT, NN, MC);
  addw_kernel<<<dim3((TT*NN+255)/256), 256, 0, stream>>>(Dmat, w);

  // --- Phase 4: closed-loop scan x_t = Acl x_{t-1} + D_t ---
  linscan_kernel<<<1, 1024, LINSCAN_LDS_BYTES, stream>>>(Acl, Dmat, w, X);

  // --- Phase 5: outputs Y = X C^T, U = V - Y K^T, costs ---
  wmma_gemm_f32<<<dim3(PP/64, TT/16), 128, 0, stream>>>(X, CT, nullptr, Ymat, TT, PP, NN);
  wmma_gemm_f32<<<dim3(MC/64, TT/16), 128, 0, stream>>>(Ymat, KTn, Vmat, Umat, TT, MC, PP);
  wmma_gemm_f32<<<dim3(PP/64, TT/16), 128, 0, stream>>>(Ymat, Q, nullptr, Z1, TT, PP, PP);
  wmma_gemm_f32<<<dim3(MC/64, TT/16), 128, 0, stream>>>(Umat, R, nullptr, Z2, TT, MC, MC);
  cost_kernel<<<dim3((TT+255)/256), 256, 0, stream>>>(Ymat, Z1, Umat, Z2, out);
}